// LocalAttention_59184649338947
// MI455X (gfx1250) — compile-verified
//
#include <hip/hip_runtime.h>
#include <math.h>

#define RNN_HID 256
#define WINDOW  128
#define SEQ_LEN 8192
#define BATCH   32
#define NEG_INF (-1e9f)

#define CTX_CHUNK 64              // rows staged into LDS per async round
#define ROW_PITCH 260             // floats per staged row (1040 B, avoids 32-way conflicts)

typedef __attribute__((ext_vector_type(2))) float v2f;
typedef __attribute__((ext_vector_type(8))) float v8f;

// ---------------------------------------------------------------------------
// Kernel 1: predicted-position head.
// One workgroup, 1024 threads = 32 waves (wave32). Each wave computes one
// 16x16 tile of tanh_out = tanh(h_t @ W_p^T) using V_WMMA_F32_16X16X4_F32
// (M=32 -> 2 M-tiles, N=256 -> 16 N-tiles, K=256 -> 64 steps of 4).
// Tiles are scaled by v_p and scattered to LDS; 32 threads then reduce each
// batch row deterministically, apply sigmoid, and emit window [start,end).
// ---------------------------------------------------------------------------
__global__ __launch_bounds__(1024) void lapos_kernel(
    const float* __restrict__ h_t, const float* __restrict__ W_p,
    const float* __restrict__ v_p, int* __restrict__ wstart,
    int* __restrict__ wend)
{
  const int COLS = RNN_HID + 1;                 // +1 pad vs LDS bank conflicts
  __shared__ float red[BATCH * (RNN_HID + 1)];

  const int tid   = threadIdx.x;
  const int wave  = tid >> 5;
  const int lane  = tid & 31;
  const int mTile = wave & 1;                   // 2 tiles cover BATCH=32
  const int nTile = wave >> 1;                  // 16 tiles cover RNN_HID=256
  const int half  = lane >> 4;                  // 0: lanes 0-15, 1: lanes 16-31
  const int l16   = lane & 15;
  const int kb    = half * 2;                   // K sub-offset per ISA layout

  const int mRow = mTile * 16 + l16;            // A-matrix row for this lane
  const int nCol = nTile * 16 + l16;            // B-matrix col for this lane

  const float* __restrict__ arow = h_t + mRow * RNN_HID;
  const float* __restrict__ brow = W_p + nCol * RNN_HID;  // B[k][n] = W_p[n][k]

  v8f c = {};
  for (int k0 = 0; k0 < RNN_HID; k0 += 4) {
    // 16x4 f32 A frag: VGPR0 = K(kb), VGPR1 = K(kb+1); 4x16 B frag likewise.
    v2f a, b;
    a[0] = arow[k0 + kb];
    a[1] = arow[k0 + kb + 1];
    b[0] = brow[k0 + kb];
    b[1] = brow[k0 + kb + 1];
    c = __builtin_amdgcn_wmma_f32_16x16x4_f32(
        /*neg_a=*/false, a, /*neg_b=*/false, b,
        /*c_mod=*/(short)0, c, /*reuse_a=*/false, /*reuse_b=*/false);
  }

  // C layout: VGPR r -> row (half*8 + r), col = lane%16 (within the tile).
  const int mBase = mTile * 16 + half * 8;
  const float vp = v_p[nCol];
#pragma unroll
  for (int r = 0; r < 8; ++r) {
    red[(mBase + r) * COLS + nCol] = tanhf(c[r]) * vp;
  }
  __syncthreads();

  if (tid < BATCH) {
    float sum = 0.f;
    const float* row = red + tid * COLS;
    for (int j = 0; j < RNN_HID; ++j) sum += row[j];     // deterministic order
    const float p = (float)SEQ_LEN / (1.f + expf(-sum)); // SEQ_LEN * sigmoid
    int pi = (int)rintf(p);                              // round-half-even
    pi = pi < 0 ? 0 : (pi > SEQ_LEN - 1 ? SEQ_LEN - 1 : pi);
    int s = pi - WINDOW; if (s < 0) s = 0;
    int e = pi + WINDOW; if (e > SEQ_LEN) e = SEQ_LEN;
    wstart[tid] = s;
    wend[tid]   = e;
  }
}

// ---------------------------------------------------------------------------
// Kernel 2: windowed softmax attention, one block per batch element.
// Out-of-window weights are exactly 0 (expf(-1e9 - max) underflows in f32),
// so we only read/compute the <=256-wide window and zero-fill the rest.
// The context pass stages window rows into LDS in 64-row chunks via
// GLOBAL_LOAD_ASYNC_TO_LDS_B128 (ASYNCcnt path), then accumulates from LDS
// with conflict-free per-lane access.
// ---------------------------------------------------------------------------
__global__ __launch_bounds__(256) void lawin_kernel(
    const float* __restrict__ enc, const float* __restrict__ w_att,
    const int* __restrict__ wstart, const int* __restrict__ wend,
    float* __restrict__ ctx_out, float* __restrict__ attn_out)
{
  __shared__ float s_watt[RNN_HID];
  __shared__ float s_w[2 * WINDOW];     // 256 == blockDim.x
  __shared__ float s_red[2 * WINDOW];
  __shared__ __align__(16) float s_tile[CTX_CHUNK * ROW_PITCH];  // ~65 KB

  const int b   = blockIdx.x;
  const int tid = threadIdx.x;
  const int start = wstart[b];
  const int end   = wend[b];
  const int W     = end - start;        // 128..256

  s_watt[tid] = w_att[tid];
  __syncthreads();

  // --- scores over the window (thread j <-> position start+j) ---
  float s = NEG_INF;
  if (tid < W) {
    const float4* row =
        (const float4*)(enc + ((size_t)b * SEQ_LEN + start + tid) * RNN_HID);
    const float4* wv = (const float4*)s_watt;
    float acc = 0.f;
#pragma unroll 4
    for (int q = 0; q < RNN_HID / 4; ++q) {
      const float4 e4 = row[q];
      const float4 w4 = wv[q];
      acc = fmaf(e4.x, w4.x, acc);
      acc = fmaf(e4.y, w4.y, acc);
      acc = fmaf(e4.z, w4.z, acc);
      acc = fmaf(e4.w, w4.w, acc);
    }
    s = acc;
  }

  // --- deterministic max reduction ---
  s_red[tid] = s;
  __syncthreads();
  for (int off = 128; off > 0; off >>= 1) {
    if (tid < off) s_red[tid] = fmaxf(s_red[tid], s_red[tid + off]);
    __syncthreads();
  }
  const float maxv = s_red[0];
  __syncthreads();

  // --- deterministic sum reduction ---
  const float e = (tid < W) ? expf(s - maxv) : 0.f;
  s_red[tid] = e;
  __syncthreads();
  for (int off = 128; off > 0; off >>= 1) {
    if (tid < off) s_red[tid] += s_red[tid + off];
    __syncthreads();
  }
  const float total = s_red[0];
  const float w = e / total;
  s_w[tid] = w;
  __syncthreads();

  // --- write attn_weights row: zeros outside window, weights inside ---
  float* __restrict__ arow = attn_out + (size_t)b * SEQ_LEN;
  for (int idx = tid; idx < SEQ_LEN; idx += 256) {
    arow[idx] = (idx >= start && idx < end) ? s_w[idx - start] : 0.f;
  }

  // --- context vector: stage window rows to LDS asynchronously, then
  //     thread h accumulates context[b][h] from LDS ---
  const unsigned long long gbase =
      (unsigned long long)(const void*)(enc +
          ((size_t)b * SEQ_LEN + start) * RNN_HID);
  // Low 32 bits of the flat shared pointer == LDS byte offset (ISA aperture mapping).
  const unsigned tile_base = (unsigned)(unsigned long long)(void*)s_tile;

  float acc = 0.f;
  for (int c0 = 0; c0 < W; c0 += CTX_CHUNK) {
    const int rows = (W - c0 < CTX_CHUNK) ? (W - c0) : CTX_CHUNK;
    // Stage rows*64 16-byte packets: row r, 16B column col.
    for (int q = tid; q < rows * (RNN_HID / 4); q += 256) {
      const int r   = q >> 6;          // row within chunk
      const int col = q & 63;          // 16-byte packet within row
      const unsigned lds_off = tile_base + (unsigned)(r * (ROW_PITCH * 4) + col * 16);
      const unsigned long long ga =
          gbase + (unsigned long long)((c0 + r) * (RNN_HID * 4) + col * 16);
      asm volatile("global_load_async_to_lds_b128 %0, %1, off"
                   :: "v"(lds_off), "v"(ga) : "memory");
    }
    asm volatile("s_wait_asynccnt 0" ::: "memory");
    __syncthreads();                    // all waves' chunks visible in LDS

    for (int j = 0; j < rows; ++j) {    // lane-consecutive LDS reads
      acc = fmaf(s_w[c0 + j], s_tile[j * ROW_PITCH + tid], acc);
    }
    __syncthreads();                    // protect tile before next stage
  }
  ctx_out[b * RNN_HID + tid] = acc;
}

extern "C" void kernel_launch(void* const* d_in, const int* in_sizes, int n_in,
                              void* d_out, int out_size, void* d_ws,
                              size_t ws_size, hipStream_t stream) {
  (void)in_sizes; (void)n_in; (void)out_size; (void)ws_size;
  const float* h_t   = (const float*)d_in[0];
  const float* enc   = (const float*)d_in[1];
  const float* W_p   = (const float*)d_in[2];
  const float* v_p   = (const float*)d_in[3];
  const float* w_att = (const float*)d_in[4];

  float* ctx_out  = (float*)d_out;                       // [32, 256]
  float* attn_out = (float*)d_out + BATCH * RNN_HID;     // [32, 8192]

  int* wstart = (int*)d_ws;
  int* wend   = wstart + BATCH;

  lapos_kernel<<<dim3(1), dim3(1024), 0, stream>>>(h_t, W_p, v_p, wstart, wend);
  lawin_kernel<<<dim3(BATCH), dim3(256), 0, stream>>>(enc, w_att, wstart, wend,
                                                      ctx_out, attn_out);
}